// UniSAGELayer_62577673502795
// MI455X (gfx1250) — compile-verified
//
#include <hip/hip_runtime.h>

typedef __attribute__((ext_vector_type(2))) float v2f;
typedef __attribute__((ext_vector_type(8))) float v8f;

#define N_NODES 100000
#define N_EDGES 20000
#define NNZ     1600000
#define D       128

// ---------------------------------------------------------------------------
// K0: zero the x_1 output region and the degree accumulator.
// ---------------------------------------------------------------------------
__global__ void zero_kernel(float* __restrict__ x1, float* __restrict__ deg) {
    int i = blockIdx.x * blockDim.x + threadIdx.x;
    if (i < N_EDGES * D) x1[i] = 0.0f;
    if (i < N_NODES)     deg[i] = 0.0f;
}

// ---------------------------------------------------------------------------
// K1: h = x_0 @ W + b via V_WMMA_F32_16X16X4_F32 (full fp32 precision).
// One 256-thread block (8 waves) computes a 16-row x 128-col slab.
// Wave w owns the 16x16 N-tile [16w, 16w+16). K=128 -> 32 WMMA ops/wave.
// W (64KB) + A-tile (8KB) staged in LDS (320KB/WGP available).
// ---------------------------------------------------------------------------
__global__ __launch_bounds__(256) void gemm_bias_kernel(
        const float* __restrict__ X, const float* __restrict__ W,
        const float* __restrict__ Bv, float* __restrict__ H) {
    __shared__ float Ws[D * D];    // 64 KB: full weight matrix
    __shared__ float As[16 * D];   //  8 KB: 16-row A tile

    const int tid  = threadIdx.x;
    const int row0 = blockIdx.x * 16;

    // Cooperative loads: W = 4096 float4 (16 per thread), A = 512 float4 (2/thread)
    const float4* Wg4 = (const float4*)W;
    float4*       Ws4 = (float4*)Ws;
#pragma unroll
    for (int i = 0; i < 16; ++i) Ws4[i * 256 + tid] = Wg4[i * 256 + tid];
    const float4* Xg4 = (const float4*)(X + (long long)row0 * D);
    float4*       As4 = (float4*)As;
    As4[tid]       = Xg4[tid];
    As4[256 + tid] = Xg4[256 + tid];
    __syncthreads();

    const int wave = tid >> 5;
    const int lane = tid & 31;
    const int m    = lane & 15;   // row within tile (A), col within tile (B/C)
    const int hi   = lane >> 4;   // half-wave selector
    const int col  = wave * 16 + m;

    // Accumulator initialized with broadcast bias: C[r][col] starts at b[col].
    const float bias = Bv[col];
    v8f c;
#pragma unroll
    for (int r = 0; r < 8; ++r) c[r] = bias;

    // ISA layouts (CDNA5 ISA):
    //  A 16x4 f32 : lanes 0-15 -> K=k,k+1 in v0,v1 ; lanes 16-31 -> K=k+2,k+3
    //  B 4x16 f32 : lanes 0-15 -> K=k (v0), k+1 (v1); lanes 16-31 -> K=k+2,k+3
    //  C 16x16    : VGPR r holds M = r + 8*hi, N = lane&15
#pragma unroll
    for (int k = 0; k < D; k += 4) {
        const int k0 = k + 2 * hi;
        v2f a;
        a[0] = As[m * D + k0];
        a[1] = As[m * D + k0 + 1];
        v2f b;
        b[0] = Ws[k0 * D + col];
        b[1] = Ws[(k0 + 1) * D + col];
        c = __builtin_amdgcn_wmma_f32_16x16x4_f32(
                /*neg_a=*/false, a, /*neg_b=*/false, b,
                /*c_mod=*/(short)0, c, /*reuse_a=*/false, /*reuse_b=*/false);
    }

#pragma unroll
    for (int r = 0; r < 8; ++r) {
        const int row = row0 + r + 8 * hi;
        H[(long long)row * D + col] = c[r];
    }
}

// ---------------------------------------------------------------------------
// K2: vertex -> edge scatter.  x_1[edge] += h[node]; deg[node] += 1.
// One thread per (nnz, 4-float chunk): 1.6M * 32 threads. h and x_1 are
// L2-resident (51MB + 10MB << 192MB L2), so gathers + atomics stay in L2.
// ---------------------------------------------------------------------------
__global__ void scatter_v2e_kernel(const float* __restrict__ H,
                                   const int* __restrict__ node_idx,
                                   const int* __restrict__ edge_idx,
                                   float* __restrict__ x1,
                                   float* __restrict__ deg) {
    unsigned gid = blockIdx.x * blockDim.x + threadIdx.x;
    if (gid >= (unsigned)NNZ * 32u) return;
    const int e  = (int)(gid >> 5);
    const int c4 = ((int)gid & 31) * 4;
    const int node = node_idx[e];
    const int edge = edge_idx[e];
    const float4 v = *(const float4*)(H + (long long)node * D + c4);
    float* dst = x1 + (long long)edge * D + c4;
    atomicAdd(dst + 0, v.x);
    atomicAdd(dst + 1, v.y);
    atomicAdd(dst + 2, v.z);
    atomicAdd(dst + 3, v.w);
    if (c4 == 0) atomicAdd(deg + node, 1.0f);
}

// ---------------------------------------------------------------------------
// K3: deg -> 1 / max(deg, 1) in place.
// ---------------------------------------------------------------------------
__global__ void invdeg_kernel(float* __restrict__ deg) {
    int i = blockIdx.x * blockDim.x + threadIdx.x;
    if (i < N_NODES) deg[i] = 1.0f / fmaxf(deg[i], 1.0f);
}

// ---------------------------------------------------------------------------
// K4: edge -> vertex scatter with fused mean normalization.
// out[node] += x_1[edge] * inv_deg[node]   (out region already holds h, so
// this directly produces x_0_out = h + m/deg; no separate m buffer needed).
// ---------------------------------------------------------------------------
__global__ void scatter_e2v_kernel(const float* __restrict__ x1,
                                   const int* __restrict__ node_idx,
                                   const int* __restrict__ edge_idx,
                                   const float* __restrict__ inv_deg,
                                   float* __restrict__ out) {
    unsigned gid = blockIdx.x * blockDim.x + threadIdx.x;
    if (gid >= (unsigned)NNZ * 32u) return;
    const int e  = (int)(gid >> 5);
    const int c4 = ((int)gid & 31) * 4;
    const int node = node_idx[e];
    const int edge = edge_idx[e];
    const float s  = inv_deg[node];
    const float4 v = *(const float4*)(x1 + (long long)edge * D + c4);
    float* dst = out + (long long)node * D + c4;
    atomicAdd(dst + 0, v.x * s);
    atomicAdd(dst + 1, v.y * s);
    atomicAdd(dst + 2, v.z * s);
    atomicAdd(dst + 3, v.w * s);
}

// ---------------------------------------------------------------------------
extern "C" void kernel_launch(void* const* d_in, const int* in_sizes, int n_in,
                              void* d_out, int out_size, void* d_ws, size_t ws_size,
                              hipStream_t stream) {
    const float* X        = (const float*)d_in[0];   // [N_NODES, D]
    const float* W        = (const float*)d_in[1];   // [D, D]
    const float* Bv       = (const float*)d_in[2];   // [D]
    const int*   node_idx = (const int*)d_in[3];     // [NNZ]
    const int*   edge_idx = (const int*)d_in[4];     // [NNZ]

    float* out = (float*)d_out;
    float* H   = out;                       // x_0_out region [N_NODES*D]
    float* x1  = out + (size_t)N_NODES * D; // x_1 region     [N_EDGES*D]
    float* deg = (float*)d_ws;              // [N_NODES] scratch (400 KB)

    // K0: zero x_1 + deg (must cover max(N_EDGES*D, N_NODES) = 2.56M elems)
    zero_kernel<<<(N_EDGES * D + 255) / 256, 256, 0, stream>>>(x1, deg);

    // K1: h = x_0 @ W + b, written straight into the output's node region
    gemm_bias_kernel<<<N_NODES / 16, 256, 0, stream>>>(X, W, Bv, H);

    // K2: x_1 = B^T h (sum) + degree counts
    const unsigned nthr = (unsigned)NNZ * 32u;
    scatter_v2e_kernel<<<(nthr + 255) / 256, 256, 0, stream>>>(
        H, node_idx, edge_idx, x1, deg);

    // K3: deg -> 1/max(deg,1)
    invdeg_kernel<<<(N_NODES + 255) / 256, 256, 0, stream>>>(deg);

    // K4: x_0_out += (B x_1) / deg
    scatter_e2v_kernel<<<(nthr + 255) / 256, 256, 0, stream>>>(
        x1, node_idx, edge_idx, deg, H);
}